// SoftProcrustesLayer_4105988735896
// MI455X (gfx1250) — compile-verified
//
#include <hip/hip_runtime.h>

// SoftProcrustes for MI455X (gfx1250).
// Single fused streaming pass over the 268MB conf matrix (~11.5us @ 23.3TB/s):
// histogram + opportunistic candidate collection; histogram-validated fast path
// with an always-enqueued (normally no-op) exact recollect fallback.
// Cross-covariance uses V_WMMA_F32_16X16X4_F32 (CDNA5 f32 matrix path).

namespace {
constexpr int B      = 16;
constexpr int N      = 2048;
constexpr int M      = 2048;
constexpr int KSEL   = 204;                 // int(2048 * 0.1)
constexpr int NBINS  = 4096;
constexpr int CAP    = 8192;                // raw candidate capacity per batch
constexpr int CAPF   = 4096;                // filtered (bin >= binT) capacity
constexpr int FIXBIN = NBINS - 2;           // opportunistic threshold: E[cnt]~2048
constexpr int BPB    = 64;                  // blocks per batch for streaming pass
constexpr int EPB    = (N * M) / BPB;       // 65536 elements per block
constexpr int PADK   = 208;                 // KSEL padded to multiple of 4
constexpr float EPSV = 1e-4f;
constexpr float MAXCOND = 100.0f;

// workspace layout (bytes); total ~1.34 MB
constexpr size_t OFF_HIST = 0;
constexpr size_t OFF_CNT  = OFF_HIST + (size_t)B * NBINS * 4;
constexpr size_t OFF_THR  = OFF_CNT  + (size_t)B * 4;
constexpr size_t OFF_FLAG = OFF_THR  + (size_t)B * 4;
constexpr size_t OFF_CV   = OFF_FLAG + (size_t)B * 4;
constexpr size_t OFF_CI   = OFF_CV   + (size_t)B * CAP * 4;
constexpr size_t OFF_SW   = OFF_CI   + (size_t)B * CAP * 4;
constexpr size_t OFF_SI   = OFF_SW   + (size_t)B * KSEL * 4;
} // namespace

typedef __attribute__((ext_vector_type(2))) float v2f;
typedef __attribute__((ext_vector_type(8))) float v8f;

__device__ __forceinline__ int bin_of(float v) {
  int bi = (int)(v * (float)NBINS);
  bi = bi < 0 ? 0 : (bi > NBINS - 1 ? NBINS - 1 : bi);
  return bi;
}

__device__ __forceinline__ float wave_sum32(float v) {
  for (int m = 16; m >= 1; m >>= 1) v += __shfl_xor(v, m, 32);
  return v;
}

// ---------------- pass 0: zero histogram + counters + flags -----------------
__global__ void k_zero(unsigned* __restrict__ hist, unsigned* __restrict__ cnt,
                       unsigned* __restrict__ flag) {
  int stride = gridDim.x * blockDim.x;
  for (int i = blockIdx.x * blockDim.x + threadIdx.x; i < B * NBINS; i += stride)
    hist[i] = 0u;
  if (blockIdx.x == 0 && threadIdx.x < B) {
    cnt[threadIdx.x] = 0u;
    flag[threadIdx.x] = 0u;
  }
}

// ---------------- pass 1 (fused, single read of conf) -----------------------
// per-batch histogram + opportunistic collection of candidates >= FIXBIN
__global__ void k_hist_collect(const float* __restrict__ conf, unsigned* __restrict__ hist,
                               unsigned* __restrict__ cnt, float* __restrict__ cv,
                               unsigned* __restrict__ ci) {
  __shared__ unsigned lh[NBINS];
  const int b = blockIdx.y;
  for (int i = threadIdx.x; i < NBINS; i += blockDim.x) lh[i] = 0u;
  __syncthreads();
  const float4* p =
      (const float4*)(conf + (size_t)b * N * M + (size_t)blockIdx.x * EPB);
  const unsigned idx0 = (unsigned)(blockIdx.x * EPB);
  const int nvec = EPB / 4;
  for (int i = threadIdx.x; i < nvec; i += blockDim.x) {
    float4 v = p[i];
    unsigned fi = idx0 + 4u * (unsigned)i;
    float vs[4] = {v.x, v.y, v.z, v.w};
#pragma unroll
    for (int c = 0; c < 4; ++c) {
      int bi = bin_of(vs[c]);
      atomicAdd(&lh[bi], 1u);
      if (bi >= FIXBIN) {
        unsigned pos = atomicAdd(&cnt[b], 1u);
        if (pos < (unsigned)CAP) {
          cv[(size_t)b * CAP + pos] = vs[c];
          ci[(size_t)b * CAP + pos] = fi + (unsigned)c;
        }
      }
    }
  }
  __syncthreads();
  for (int i = threadIdx.x; i < NBINS; i += blockDim.x)
    if (lh[i]) atomicAdd(&hist[b * NBINS + i], lh[i]);
}

// ---------------- pass 2: exact threshold bin + fast-path validation --------
__global__ void k_thresh(const unsigned* __restrict__ hist, unsigned* __restrict__ thr,
                         unsigned* __restrict__ cnt, unsigned* __restrict__ flag) {
  int b = threadIdx.x;
  if (b < B) {
    unsigned cum = 0;
    int bin = 0;
    for (int i = NBINS - 1; i >= 0; --i) {
      cum += hist[b * NBINS + i];
      if (cum >= (unsigned)KSEL) { bin = i; break; }
    }
    thr[b] = (unsigned)bin;
    // fast path valid iff every element with bin >= binT was stored:
    // binT must be inside the opportunistic range and no appends were dropped.
    bool ok = (bin >= FIXBIN) && (cnt[b] <= (unsigned)CAP);
    flag[b] = ok ? 0u : 1u;
    if (!ok) cnt[b] = 0u;   // fallback recollect refills from scratch
  }
}

// ---------------- pass 3: exact recollect (normally a no-op) ----------------
__global__ void k_recollect(const float* __restrict__ conf, const unsigned* __restrict__ thr,
                            const unsigned* __restrict__ flag, unsigned* __restrict__ cnt,
                            float* __restrict__ cv, unsigned* __restrict__ ci) {
  const int b = blockIdx.y;
  if (flag[b] == 0u) return;          // fast path already holds a valid superset
  const int binT = (int)thr[b];
  const float4* p =
      (const float4*)(conf + (size_t)b * N * M + (size_t)blockIdx.x * EPB);
  const unsigned idx0 = (unsigned)(blockIdx.x * EPB);
  const int nvec = EPB / 4;
  for (int i = threadIdx.x; i < nvec; i += blockDim.x) {
    float4 v = p[i];
    unsigned fi = idx0 + 4u * (unsigned)i;
    float vs[4] = {v.x, v.y, v.z, v.w};
#pragma unroll
    for (int c = 0; c < 4; ++c) {
      if (bin_of(vs[c]) >= binT) {
        unsigned pos = atomicAdd(&cnt[b], 1u);
        if (pos < (unsigned)CAP) {
          cv[(size_t)b * CAP + pos] = vs[c];
          ci[(size_t)b * CAP + pos] = fi + (unsigned)c;
        }
      }
    }
  }
}

// ---------------- pass 4: filter to bin>=binT, then exact top-KSEL ----------
// rank tie-break matches lax.top_k: value desc, then lower flat index first.
__global__ void k_select(const float* __restrict__ cv, const unsigned* __restrict__ ci,
                         const unsigned* __restrict__ cnt, const unsigned* __restrict__ thr,
                         float* __restrict__ selw, unsigned* __restrict__ seli) {
  __shared__ float fv[CAPF];
  __shared__ unsigned fidx[CAPF];
  __shared__ unsigned fc;
  const int b = blockIdx.x;
  const int binT = (int)thr[b];
  unsigned c = cnt[b];
  if (c > (unsigned)CAP) c = (unsigned)CAP;
  if (threadIdx.x == 0) fc = 0u;
  __syncthreads();
  // compact candidates that pass the exact threshold into LDS
  for (unsigned i = threadIdx.x; i < c; i += blockDim.x) {
    float v = cv[(size_t)b * CAP + i];
    if (bin_of(v) >= binT) {
      unsigned pos = atomicAdd(&fc, 1u);
      if (pos < (unsigned)CAPF) {
        fv[pos] = v;
        fidx[pos] = ci[(size_t)b * CAP + i];
      }
    }
  }
  __syncthreads();
  unsigned n = fc;
  if (n > (unsigned)CAPF) n = (unsigned)CAPF;
  for (unsigned i = threadIdx.x; i < n; i += blockDim.x) {
    const float vi = fv[i];
    const unsigned myidx = fidx[i];
    unsigned rank = 0;
    for (unsigned j = 0; j < n; ++j) {
      float vj = fv[j];
      if (vj > vi) rank++;
      else if (vj == vi && fidx[j] < myidx) rank++;
    }
    if (rank < (unsigned)KSEL) {
      selw[(size_t)b * KSEL + rank] = vi;
      seli[(size_t)b * KSEL + rank] = myidx;
    }
  }
}

// ---------------- pass 5: weighted Procrustes, WMMA covariance, 3x3 SVD -----
__global__ void k_procrustes(const float* __restrict__ src, const float* __restrict__ tgt,
                             const float* __restrict__ selw, const unsigned* __restrict__ seli,
                             float* __restrict__ out) {
  __shared__ float Xl[PADK][3];
  __shared__ float Yl[PADK][3];
  __shared__ float Wl[PADK];
  __shared__ float wXc[PADK][3];   // w_norm * (X - meanX)
  __shared__ float Ycs[PADK][3];   // (Y - meanY)
  __shared__ float Smat[9];

  const int b = blockIdx.x;
  const int lane = threadIdx.x;   // 32 threads = one wave

  // gather selected pairs (zero-pad to PADK)
  for (int e = lane; e < PADK; e += 32) {
    float w = 0.f, x0 = 0.f, x1 = 0.f, x2 = 0.f, y0 = 0.f, y1 = 0.f, y2 = 0.f;
    if (e < KSEL) {
      w = selw[(size_t)b * KSEL + e];
      unsigned idx = seli[(size_t)b * KSEL + e];
      unsigned is = idx >> 11;       // / M
      unsigned it = idx & 2047u;     // % M
      const float* xp = src + ((size_t)b * N + is) * 3;
      const float* yp = tgt + ((size_t)b * M + it) * 3;
      x0 = xp[0]; x1 = xp[1]; x2 = xp[2];
      y0 = yp[0]; y1 = yp[1]; y2 = yp[2];
    }
    Wl[e] = w;
    Xl[e][0] = x0; Xl[e][1] = x1; Xl[e][2] = x2;
    Yl[e][0] = y0; Yl[e][1] = y1; Yl[e][2] = y2;
  }
  __syncthreads();

  // W1 = sum |w|
  float pw = 0.f;
  for (int e = lane; e < KSEL; e += 32) pw += fabsf(Wl[e]);
  const float W1e = wave_sum32(pw) + EPSV;

  // weighted means
  float sx0 = 0.f, sx1 = 0.f, sx2 = 0.f, sy0 = 0.f, sy1 = 0.f, sy2 = 0.f;
  for (int e = lane; e < KSEL; e += 32) {
    float wn = Wl[e] / W1e;
    sx0 += wn * Xl[e][0]; sx1 += wn * Xl[e][1]; sx2 += wn * Xl[e][2];
    sy0 += wn * Yl[e][0]; sy1 += wn * Yl[e][1]; sy2 += wn * Yl[e][2];
  }
  sx0 = wave_sum32(sx0); sx1 = wave_sum32(sx1); sx2 = wave_sum32(sx2);
  sy0 = wave_sum32(sy0); sy1 = wave_sum32(sy1); sy2 = wave_sum32(sy2);

  // centered arrays (padded rows have w=0 -> wXc = 0 -> no WMMA contribution)
  for (int e = lane; e < PADK; e += 32) {
    float wn = Wl[e] / W1e;
    wXc[e][0] = wn * (Xl[e][0] - sx0);
    wXc[e][1] = wn * (Xl[e][1] - sx1);
    wXc[e][2] = wn * (Xl[e][2] - sx2);
    Ycs[e][0] = Yl[e][0] - sy0;
    Ycs[e][1] = Yl[e][1] - sy1;
    Ycs[e][2] = Yl[e][2] - sy2;
  }
  __syncthreads();

  // Sxy (3x3) = Yc^T [3 x K] @ wXc [K x 3] via V_WMMA_F32_16X16X4_F32.
  // A 16x4 layout: lanes 0-15 -> M=lane, v0=K0,v1=K1; lanes 16-31 -> K2,K3.
  // B 4x16 layout: vr: lanes 0-15 -> K=r, lanes 16-31 -> K=r+2; N=lane&15.
  v8f acc = {0.f, 0.f, 0.f, 0.f, 0.f, 0.f, 0.f, 0.f};
  const int mrow = lane & 15;
  const int koff = (lane >> 4) * 2;
  const bool valid = (mrow < 3);
  const int mc = valid ? mrow : 0;
  for (int kb = 0; kb < PADK; kb += 4) {
    const int n0 = kb + koff;
    const int n1 = n0 + 1;
    float a0 = Ycs[n0][mc], a1 = Ycs[n1][mc];
    float b0 = wXc[n0][mc], b1 = wXc[n1][mc];
    v2f av, bv;
    av.x = valid ? a0 : 0.f;
    av.y = valid ? a1 : 0.f;
    bv.x = valid ? b0 : 0.f;
    bv.y = valid ? b1 : 0.f;
    acc = __builtin_amdgcn_wmma_f32_16x16x4_f32(false, av, false, bv,
                                                (short)0, acc, false, false);
  }
  // D layout: VGPR i, lanes 0-15 -> row i, column = lane
  if (lane < 3) {
    Smat[0 * 3 + lane] = acc[0];
    Smat[1 * 3 + lane] = acc[1];
    Smat[2 * 3 + lane] = acc[2];
  }
  __syncthreads();

  if (lane == 0) {
    float A9[9];
#pragma unroll
    for (int i = 0; i < 9; ++i) A9[i] = Smat[i];

    // Jacobi eigendecomposition of A^T A -> V, lambda
    float Bs[9];
    for (int i = 0; i < 3; ++i)
      for (int j = 0; j < 3; ++j) {
        float s = 0.f;
        for (int k2 = 0; k2 < 3; ++k2) s += A9[k2 * 3 + i] * A9[k2 * 3 + j];
        Bs[i * 3 + j] = s;
      }
    float V9[9] = {1.f, 0.f, 0.f, 0.f, 1.f, 0.f, 0.f, 0.f, 1.f};
    const int PQ[3][2] = {{0, 1}, {0, 2}, {1, 2}};
    for (int sweep = 0; sweep < 12; ++sweep) {
      for (int r = 0; r < 3; ++r) {
        int p = PQ[r][0], q = PQ[r][1];
        float apq = Bs[p * 3 + q];
        if (fabsf(apq) > 1e-30f) {
          float app = Bs[p * 3 + p], aqq = Bs[q * 3 + q];
          float tau = (aqq - app) / (2.0f * apq);
          float tsg = (tau >= 0.f) ? 1.f : -1.f;
          float tt = tsg / (fabsf(tau) + sqrtf(1.f + tau * tau));
          float cc = 1.0f / sqrtf(1.f + tt * tt);
          float ss = tt * cc;
          for (int k2 = 0; k2 < 3; ++k2) {
            float bkp = Bs[k2 * 3 + p], bkq = Bs[k2 * 3 + q];
            Bs[k2 * 3 + p] = cc * bkp - ss * bkq;
            Bs[k2 * 3 + q] = ss * bkp + cc * bkq;
          }
          for (int k2 = 0; k2 < 3; ++k2) {
            float bpk = Bs[p * 3 + k2], bqk = Bs[q * 3 + k2];
            Bs[p * 3 + k2] = cc * bpk - ss * bqk;
            Bs[q * 3 + k2] = ss * bpk + cc * bqk;
          }
          for (int k2 = 0; k2 < 3; ++k2) {
            float vkp = V9[k2 * 3 + p], vkq = V9[k2 * 3 + q];
            V9[k2 * 3 + p] = cc * vkp - ss * vkq;
            V9[k2 * 3 + q] = ss * vkp + cc * vkq;
          }
        }
      }
    }
    float lam[3] = {Bs[0], Bs[4], Bs[8]};
    int ord[3] = {0, 1, 2};
    for (int i = 0; i < 2; ++i)
      for (int j = i + 1; j < 3; ++j)
        if (lam[ord[j]] > lam[ord[i]]) { int t = ord[i]; ord[i] = ord[j]; ord[j] = t; }
    float Dg[3], Vc[9];
    for (int c2 = 0; c2 < 3; ++c2) {
      Dg[c2] = sqrtf(fmaxf(lam[ord[c2]], 0.f));
      for (int rI = 0; rI < 3; ++rI) Vc[rI * 3 + c2] = V9[rI * 3 + ord[c2]];
    }
    // U columns: u_i = A v_i / sigma_i ; u2 = u0 x u1 (R invariant to sign choices)
    float Uc[9];
    for (int c2 = 0; c2 < 2; ++c2) {
      float u[3];
      for (int rI = 0; rI < 3; ++rI)
        u[rI] = A9[rI * 3 + 0] * Vc[0 * 3 + c2] + A9[rI * 3 + 1] * Vc[1 * 3 + c2] +
                A9[rI * 3 + 2] * Vc[2 * 3 + c2];
      if (c2 == 1) {
        float d0 = u[0] * Uc[0] + u[1] * Uc[3] + u[2] * Uc[6];
        u[0] -= d0 * Uc[0]; u[1] -= d0 * Uc[3]; u[2] -= d0 * Uc[6];
      }
      float nrm = sqrtf(u[0] * u[0] + u[1] * u[1] + u[2] * u[2]);
      float inv = (nrm > 0.f) ? 1.f / nrm : 0.f;
      Uc[0 * 3 + c2] = u[0] * inv;
      Uc[1 * 3 + c2] = u[1] * inv;
      Uc[2 * 3 + c2] = u[2] * inv;
    }
    Uc[0 * 3 + 2] = Uc[3] * Uc[7] - Uc[6] * Uc[4];
    Uc[1 * 3 + 2] = Uc[6] * Uc[1] - Uc[0] * Uc[7];
    Uc[2 * 3 + 2] = Uc[0] * Uc[4] - Uc[3] * Uc[1];

    auto det3 = [](const float* Mx) {
      return Mx[0] * (Mx[4] * Mx[8] - Mx[5] * Mx[7]) -
             Mx[1] * (Mx[3] * Mx[8] - Mx[5] * Mx[6]) +
             Mx[2] * (Mx[3] * Mx[7] - Mx[4] * Mx[6]);
    };
    float d = det3(Uc) * det3(Vc);

    float R[9];
    for (int i = 0; i < 3; ++i)
      for (int j = 0; j < 3; ++j)
        R[i * 3 + j] = Uc[i * 3 + 0] * Vc[j * 3 + 0] + Uc[i * 3 + 1] * Vc[j * 3 + 1] +
                       d * Uc[i * 3 + 2] * Vc[j * 3 + 2];
    float mX[3] = {sx0, sx1, sx2}, mY[3] = {sy0, sy1, sy2};
    float tv[3];
    for (int i = 0; i < 3; ++i)
      tv[i] = mY[i] - (R[i * 3 + 0] * mX[0] + R[i * 3 + 1] * mX[1] + R[i * 3 + 2] * mX[2]);

    float cond = Dg[0] / Dg[2];
    bool ok = (cond < MAXCOND);

    // output layout: R[B*9] | t[B*3] | R_forwd[B*9] | t_forwd[B*3] | cond[B] | mask[B]
    float* Ro = out + (size_t)b * 9;
    float* To = out + 144 + (size_t)b * 3;
    float* Rf = out + 192 + (size_t)b * 9;
    float* Tf = out + 336 + (size_t)b * 3;
    for (int i = 0; i < 9; ++i) {
      Ro[i] = R[i];
      float eye = (i % 4 == 0) ? 1.f : 0.f;   // i = 0,4,8 diagonal
      Rf[i] = ok ? R[i] : eye;
    }
    for (int i = 0; i < 3; ++i) {
      To[i] = tv[i];
      Tf[i] = ok ? tv[i] : 0.f;
    }
    out[384 + b] = cond;
    out[400 + b] = ok ? 1.f : 0.f;
  }
}

extern "C" void kernel_launch(void* const* d_in, const int* in_sizes, int n_in,
                              void* d_out, int out_size, void* d_ws, size_t ws_size,
                              hipStream_t stream) {
  (void)in_sizes; (void)n_in; (void)out_size; (void)ws_size;
  const float* conf = (const float*)d_in[0];
  const float* src  = (const float*)d_in[1];
  const float* tgt  = (const float*)d_in[2];
  float* out = (float*)d_out;

  char* ws = (char*)d_ws;
  unsigned* hist = (unsigned*)(ws + OFF_HIST);
  unsigned* cnt  = (unsigned*)(ws + OFF_CNT);
  unsigned* thr  = (unsigned*)(ws + OFF_THR);
  unsigned* flag = (unsigned*)(ws + OFF_FLAG);
  float*    cv   = (float*)(ws + OFF_CV);
  unsigned* ci   = (unsigned*)(ws + OFF_CI);
  float*    selw = (float*)(ws + OFF_SW);
  unsigned* seli = (unsigned*)(ws + OFF_SI);

  k_zero<<<64, 256, 0, stream>>>(hist, cnt, flag);
  k_hist_collect<<<dim3(BPB, B), 256, 0, stream>>>(conf, hist, cnt, cv, ci);
  k_thresh<<<1, 32, 0, stream>>>(hist, thr, cnt, flag);
  k_recollect<<<dim3(BPB, B), 256, 0, stream>>>(conf, thr, flag, cnt, cv, ci);
  k_select<<<B, 256, 0, stream>>>(cv, ci, cnt, thr, selw, seli);
  k_procrustes<<<B, 32, 0, stream>>>(src, tgt, selw, seli, out);
}